// Net_90074054132247
// MI455X (gfx1250) — compile-verified
//
#include <hip/hip_runtime.h>
#include <hip/hip_bf16.h>

typedef __attribute__((ext_vector_type(2))) float v2f;
typedef __attribute__((ext_vector_type(8))) float v8f;

#define NNODES 4096
#define NEDGES 131072
#define C_IN   20
#define C_HID  64
#define C_OUT  20
// top-k sizes: ceil(0.8*n) chain
#define K1 3277
#define K2 2622
#define K3 2098
// padded (multiple of 128) leading dims
#define P0 4096
#define P1 3328
#define P2 2688
#define P3 2176

// ---------------------------------------------------------------- utilities

__global__ void fill_zero_f32(float* __restrict__ p, size_t n) {
  size_t i = (size_t)blockIdx.x * blockDim.x + threadIdx.x;
  size_t st = (size_t)gridDim.x * blockDim.x;
  for (; i < n; i += st) p[i] = 0.0f;
}

__global__ void build_a0_edges(const int* __restrict__ ei, float* __restrict__ A0) {
  int e = blockIdx.x * blockDim.x + threadIdx.x;
  if (e >= NEDGES) return;
  int s = ei[e];
  int d = ei[NEDGES + e];
  if (s != d) atomicAdd(&A0[(size_t)s * P0 + d], 1.0f);
}

__global__ void add_eye(float* __restrict__ A) {
  int i = blockIdx.x * blockDim.x + threadIdx.x;
  if (i < NNODES) A[(size_t)i * P0 + i] += 1.0f;
}

__global__ void zero_diag(float* __restrict__ C, int np) {
  int i = blockIdx.x * blockDim.x + threadIdx.x;
  if (i < np) C[(size_t)i * np + i] = 0.0f;
}

__global__ void copy_f32(const float* __restrict__ s, float* __restrict__ d, int n) {
  int i = blockIdx.x * blockDim.x + threadIdx.x;
  if (i < n) d[i] = s[i];
}

// --------------------------------------------------- augment: C = A1 @ A1
// A1[i][j] = (i==j) ? 1 : A[i][j]   (remove self loops + add self loops)
// 8 waves/block; block tile 128x64; wave tile 32x32; BK=32; fp32 WMMA 16x16x4.
__global__ __launch_bounds__(256) void augment_gemm(const float* __restrict__ A,
                                                    float* __restrict__ C, int np) {
  __shared__ float As_[128][33];
  __shared__ float Bs_[32][64];
  const int tid  = threadIdx.x;
  const int lane = tid & 31;
  const int wave = tid >> 5;
  const int bm = blockIdx.x * 128;
  const int bn = blockIdx.y * 64;
  const int wm = (wave >> 1) * 32;  // 0,32,64,96
  const int wn = (wave & 1) * 32;   // 0,32
  const int hk  = ((lane >> 4) & 1) * 2;  // K group: lanes 0-15 -> K0/K1, 16-31 -> K2/K3
  const int l16 = lane & 15;
  v8f acc[2][2] = {};

  for (int k0 = 0; k0 < np; k0 += 32) {
    // stage A tile (rows bm..+127, cols k0..+31) with self-loop transform
#pragma unroll
    for (int t = 0; t < 16; ++t) {
      int l = tid + t * 256;
      int r = l >> 5, c = l & 31;
      int gr = bm + r, gc = k0 + c;
      float v = A[(size_t)gr * np + gc];
      if (gr == gc) v = 1.0f;
      As_[r][c] = v;
    }
    // stage B tile (rows k0..+31, cols bn..+63)
#pragma unroll
    for (int t = 0; t < 8; ++t) {
      int l = tid + t * 256;
      int r = l >> 6, c = l & 63;
      int gr = k0 + r, gc = bn + c;
      float v = A[(size_t)gr * np + gc];
      if (gr == gc) v = 1.0f;
      Bs_[r][c] = v;
    }
    __syncthreads();
#pragma unroll
    for (int kk = 0; kk < 32; kk += 4) {
      v2f af[2], bf[2];
#pragma unroll
      for (int a = 0; a < 2; a++) {
        af[a].x = As_[wm + a * 16 + l16][kk + hk];
        af[a].y = As_[wm + a * 16 + l16][kk + hk + 1];
      }
#pragma unroll
      for (int b = 0; b < 2; b++) {
        bf[b].x = Bs_[kk + hk][wn + b * 16 + l16];
        bf[b].y = Bs_[kk + hk + 1][wn + b * 16 + l16];
      }
#pragma unroll
      for (int a = 0; a < 2; a++)
#pragma unroll
        for (int b = 0; b < 2; b++)
          acc[a][b] = __builtin_amdgcn_wmma_f32_16x16x4_f32(
              false, af[a], false, bf[b], (short)0, acc[a][b], false, false);
    }
    __syncthreads();
  }
  const int rh = (lane >> 4) * 8;
#pragma unroll
  for (int a = 0; a < 2; a++)
#pragma unroll
    for (int b = 0; b < 2; b++)
#pragma unroll
      for (int q = 0; q < 8; q++) {
        int row = bm + wm + a * 16 + q + rh;
        int col = bn + wn + b * 16 + l16;
        C[(size_t)row * np + col] = acc[a][b][q];
      }
}

// ------------------------------------------------------------- GCN pieces
// deg over columns of (A + diag(fix)); dis = 1/sqrt(deg)
__global__ void deg_fix(const float* __restrict__ A, float* __restrict__ dis,
                        float* __restrict__ fix, int n, int np) {
  int c = blockIdx.x * blockDim.x + threadIdx.x;
  if (c >= n) return;
  float s = 0.0f;
  for (int r = 0; r < n; r++) s += A[(size_t)r * np + c];
  float fx = (A[(size_t)c * np + c] == 0.0f) ? 2.0f : 0.0f;
  float deg = s + fx;
  dis[c] = (deg > 0.0f) ? (1.0f / sqrtf(deg)) : 0.0f;
  fix[c] = fx;
}

// y2[r,f] = dis[r] * (x @ W)[r,f], zero-padded to np rows / ldy cols
__global__ void feat_matmul(const float* __restrict__ x, const float* __restrict__ W,
                            const float* __restrict__ dis, float* __restrict__ y2,
                            int n, int np, int cin, int cout, int ldy) {
  int idx = blockIdx.x * blockDim.x + threadIdx.x;
  int r = idx / ldy, f = idx % ldy;
  if (r >= np) return;
  float v = 0.0f;
  if (r < n && f < cout) {
    float a = 0.0f;
    for (int c = 0; c < cin; c++) a += x[r * cin + c] * W[c * cout + f];
    v = dis[r] * a;
  }
  y2[idx] = v;
}

// out[c,f] = dis[c]*( (A^T y2)[c,f] + fix[c]*y2[c,f] ) + b[f]  (opt. relu)
// WMMA: M=c (16/wave), N=f (NF frags), K=r. A fragment read transposed.
template <int NF>
__global__ __launch_bounds__(256) void agg_gemm(
    const float* __restrict__ A, const float* __restrict__ y2,
    const float* __restrict__ dis, const float* __restrict__ fix,
    const float* __restrict__ bias, float* __restrict__ out,
    int n, int np, int cout, int ldo, int dorelu) {
  const int tid = threadIdx.x, lane = tid & 31, wave = tid >> 5;
  const int c0 = blockIdx.x * 128 + wave * 16;
  const int ldy = NF * 16;
  const int hk = ((lane >> 4) & 1) * 2, l16 = lane & 15;
  v8f acc[NF] = {};
  for (int r0 = 0; r0 < np; r0 += 4) {
    v2f af;
    af.x = A[(size_t)(r0 + hk) * np + c0 + l16];
    af.y = A[(size_t)(r0 + hk + 1) * np + c0 + l16];
#pragma unroll
    for (int j = 0; j < NF; j++) {
      v2f bf;
      bf.x = y2[(r0 + hk) * ldy + j * 16 + l16];
      bf.y = y2[(r0 + hk + 1) * ldy + j * 16 + l16];
      acc[j] = __builtin_amdgcn_wmma_f32_16x16x4_f32(
          false, af, false, bf, (short)0, acc[j], false, false);
    }
  }
  const int rh = (lane >> 4) * 8;
#pragma unroll
  for (int j = 0; j < NF; j++)
#pragma unroll
    for (int q = 0; q < 8; q++) {
      int c = c0 + q + rh;
      int f = j * 16 + l16;
      if (c < n && f < cout) {
        float val = dis[c] * (acc[j][q] + fix[c] * y2[c * ldy + f]) + bias[f];
        if (dorelu) val = fmaxf(val, 0.0f);
        out[(size_t)c * ldo + f] = val;
      }
    }
}

// ------------------------------------------------------------ top-k pooling
__global__ void score_keys(const float* __restrict__ x, const float* __restrict__ p,
                           int n, float* __restrict__ scores,
                           unsigned long long* __restrict__ keys) {
  int i = blockIdx.x * blockDim.x + threadIdx.x;
  if (i >= 4096) return;
  unsigned long long key = 0ull;
  float s = 0.0f;
  if (i < n) {
    float pn = 0.0f, d = 0.0f;
    for (int c = 0; c < C_HID; c++) pn += p[c] * p[c];
    for (int c = 0; c < C_HID; c++) d += x[i * C_HID + c] * p[c];
    s = tanhf(d / sqrtf(pn));
    unsigned u = __float_as_uint(s);
    u = (u & 0x80000000u) ? ~u : (u | 0x80000000u);  // monotonic map
    key = ((unsigned long long)u << 32) |
          (unsigned long long)(0xFFFFFFFFu - (unsigned)i);  // low-index ties first
  }
  scores[i] = s;
  keys[i] = key;
}

__global__ __launch_bounds__(1024) void bitonic4096(unsigned long long* __restrict__ keys) {
  __shared__ unsigned long long s[4096];
  int t = threadIdx.x;
  for (int i = t; i < 4096; i += 1024) s[i] = keys[i];
  __syncthreads();
  for (int k = 2; k <= 4096; k <<= 1) {
    for (int j = k >> 1; j > 0; j >>= 1) {
      for (int i = t; i < 4096; i += 1024) {
        int l = i ^ j;
        if (l > i) {
          unsigned long long a = s[i], b = s[l];
          bool dirDesc = ((i & k) == 0);
          bool sw = dirDesc ? (a < b) : (a > b);
          if (sw) { s[i] = b; s[l] = a; }
        }
      }
      __syncthreads();
    }
  }
  for (int i = t; i < 4096; i += 1024) keys[i] = s[i];
}

__global__ void extract_topk(const unsigned long long* __restrict__ keys,
                             const float* __restrict__ scores, int k,
                             int* __restrict__ perm, float* __restrict__ sval) {
  int i = blockIdx.x * blockDim.x + threadIdx.x;
  if (i >= k) return;
  unsigned idx = 0xFFFFFFFFu - (unsigned)(keys[i] & 0xFFFFFFFFull);
  perm[i] = (int)idx;
  sval[i] = scores[idx];
}

__global__ void pool_gather_x(const float* __restrict__ xin, const int* __restrict__ perm,
                              const float* __restrict__ sval, float* __restrict__ xout, int k) {
  int idx = blockIdx.x * blockDim.x + threadIdx.x;
  if (idx >= k * C_HID) return;
  int i = idx >> 6, f = idx & 63;
  xout[idx] = xin[perm[i] * C_HID + f] * sval[i];
}

__global__ void pool_gather_A(const float* __restrict__ B, int ldB,
                              const int* __restrict__ perm, float* __restrict__ Aout,
                              int ldA, int k) {
  size_t idx = (size_t)blockIdx.x * blockDim.x + threadIdx.x;
  size_t tot = (size_t)ldA * ldA;
  if (idx >= tot) return;
  int i = (int)(idx / ldA), j = (int)(idx % ldA);
  float v = 0.0f;
  if (i < k && j < k) v = B[(size_t)perm[i] * ldB + perm[j]];
  Aout[idx] = v;
}

__global__ void scatter_add_x(float* __restrict__ dst, const int* __restrict__ perm,
                              const float* __restrict__ src, int k) {
  int idx = blockIdx.x * blockDim.x + threadIdx.x;
  if (idx >= k * C_HID) return;
  int i = idx >> 6, f = idx & 63;
  dst[perm[i] * C_HID + f] += src[idx];
}

// -------------------------------------------------------------- launcher

extern "C" void kernel_launch(void* const* d_in, const int* in_sizes, int n_in,
                              void* d_out, int out_size, void* d_ws, size_t ws_size,
                              hipStream_t stream) {
  (void)in_sizes; (void)n_in; (void)out_size; (void)ws_size;
  const float* x_in = (const float*)d_in[0];
  const int*   eidx = (const int*)d_in[1];
  const float* W0 = (const float*)d_in[2];  const float* b0 = (const float*)d_in[3];
  const float* W1 = (const float*)d_in[4];  const float* b1 = (const float*)d_in[5];
  const float* W2 = (const float*)d_in[6];  const float* b2 = (const float*)d_in[7];
  const float* W3 = (const float*)d_in[8];  const float* b3 = (const float*)d_in[9];
  const float* p1 = (const float*)d_in[10]; const float* p2 = (const float*)d_in[11];
  const float* p3 = (const float*)d_in[12];
  const float* U0 = (const float*)d_in[13]; const float* c0 = (const float*)d_in[14];
  const float* U1 = (const float*)d_in[15]; const float* c1 = (const float*)d_in[16];
  const float* U2 = (const float*)d_in[17]; const float* c2 = (const float*)d_in[18];
  float* out = (float*)d_out;

  // carve workspace (floats)
  float* w = (float*)d_ws;
  size_t off = 0;
  auto carve = [&](size_t nf) { float* p = w + off; off += nf; return p; };
  float* A0b = carve((size_t)P0 * P0);
  float* Bb  = carve((size_t)P0 * P0);
  float* A1b = carve((size_t)P1 * P1);
  float* A2b = carve((size_t)P2 * P2);
  float* A3b = carve((size_t)P3 * P3);
  float* X0  = carve((size_t)P0 * C_HID);
  float* X1  = carve((size_t)P1 * C_HID);
  float* X2  = carve((size_t)P2 * C_HID);
  float* X3  = carve((size_t)P3 * C_HID);
  float* XW  = carve((size_t)P0 * C_HID);
  float* XU  = carve((size_t)P0 * C_HID);
  float* Y2  = carve((size_t)P0 * C_HID);
  float* scores = carve(4096);
  unsigned long long* keys = (unsigned long long*)carve(8192);
  int* perm0 = (int*)carve(4096);
  int* perm1 = (int*)carve(4096);
  int* perm2 = (int*)carve(4096);
  float* sv0 = carve(4096);
  float* sv1 = carve(4096);
  float* sv2 = carve(4096);
  float* dis = carve(4096);
  float* fix = carve(4096);

  auto gcn = [&](const float* A, int n, int np, const float* xin, int cin,
                 const float* W, const float* b, int cout, int nf,
                 float* o, int ldo, int relu) {
    int ldy = nf * 16;
    deg_fix<<<dim3((n + 255) / 256), dim3(256), 0, stream>>>(A, dis, fix, n, np);
    feat_matmul<<<dim3((np * ldy + 255) / 256), dim3(256), 0, stream>>>(
        xin, W, dis, Y2, n, np, cin, cout, ldy);
    if (nf == 4)
      agg_gemm<4><<<dim3(np / 128), dim3(256), 0, stream>>>(
          A, Y2, dis, fix, b, o, n, np, cout, ldo, relu);
    else
      agg_gemm<2><<<dim3(np / 128), dim3(256), 0, stream>>>(
          A, Y2, dis, fix, b, o, n, np, cout, ldo, relu);
  };
  auto pool = [&](const float* A, int np, const float* xcur, int ncur,
                  const float* pv, int k, int* perm, float* sval,
                  float* Anew, int npnew) {
    augment_gemm<<<dim3(np / 128, np / 64), dim3(256), 0, stream>>>(A, Bb, np);
    zero_diag<<<dim3((np + 255) / 256), dim3(256), 0, stream>>>(Bb, np);
    score_keys<<<dim3(16), dim3(256), 0, stream>>>(xcur, pv, ncur, scores, keys);
    bitonic4096<<<dim3(1), dim3(1024), 0, stream>>>(keys);
    extract_topk<<<dim3((k + 255) / 256), dim3(256), 0, stream>>>(keys, scores, k, perm, sval);
    pool_gather_x<<<dim3((k * C_HID + 255) / 256), dim3(256), 0, stream>>>(xcur, perm, sval, XW, k);
    size_t tot = (size_t)npnew * npnew;
    pool_gather_A<<<dim3((unsigned)((tot + 255) / 256)), dim3(256), 0, stream>>>(
        Bb, np, perm, Anew, npnew, k);
  };

  // ---- build A0
  fill_zero_f32<<<dim3(8192), dim3(256), 0, stream>>>(A0b, (size_t)P0 * P0);
  build_a0_edges<<<dim3(NEDGES / 256), dim3(256), 0, stream>>>(eidx, A0b);
  add_eye<<<dim3(NNODES / 256), dim3(256), 0, stream>>>(A0b);

  // ---- down path
  gcn(A0b, NNODES, P0, x_in, C_IN, W0, b0, C_HID, 4, X0, C_HID, 1);
  pool(A0b, P0, X0, NNODES, p1, K1, perm0, sv0, A1b, P1);
  gcn(A1b, K1, P1, XW, C_HID, W1, b1, C_HID, 4, X1, C_HID, 1);
  pool(A1b, P1, X1, K1, p2, K2, perm1, sv1, A2b, P2);
  gcn(A2b, K2, P2, XW, C_HID, W2, b2, C_HID, 4, X2, C_HID, 1);
  pool(A2b, P2, X2, K2, p3, K3, perm2, sv2, A3b, P3);
  gcn(A3b, K3, P3, XW, C_HID, W3, b3, C_HID, 4, X3, C_HID, 1);

  // ---- up path (sum_res unpool then conv)
  copy_f32<<<dim3((K2 * C_HID + 255) / 256), dim3(256), 0, stream>>>(X2, XW, K2 * C_HID);
  scatter_add_x<<<dim3((K3 * C_HID + 255) / 256), dim3(256), 0, stream>>>(XW, perm2, X3, K3);
  gcn(A2b, K2, P2, XW, C_HID, U0, c0, C_HID, 4, XU, C_HID, 1);

  copy_f32<<<dim3((K1 * C_HID + 255) / 256), dim3(256), 0, stream>>>(X1, XW, K1 * C_HID);
  scatter_add_x<<<dim3((K2 * C_HID + 255) / 256), dim3(256), 0, stream>>>(XW, perm1, XU, K2);
  gcn(A1b, K1, P1, XW, C_HID, U1, c1, C_HID, 4, XU, C_HID, 1);

  copy_f32<<<dim3((NNODES * C_HID + 255) / 256), dim3(256), 0, stream>>>(X0, XW, NNODES * C_HID);
  scatter_add_x<<<dim3((K1 * C_HID + 255) / 256), dim3(256), 0, stream>>>(XW, perm0, XU, K1);
  gcn(A0b, NNODES, P0, XW, C_HID, U2, c2, C_OUT, 2, out, C_OUT, 0);
}